// CombinedEmbedding_936302871087
// MI455X (gfx1250) — compile-verified
//
#include <hip/hip_runtime.h>

// ---------------------------------------------------------------------------
// CombinedEmbedding for MI455X (gfx1250, wave32).
//
// out[p, e] = W[e, idx[p]] + pos_emb[s(p), e] + fmap_emb[f(p), e]
// where idx[p] = argmax of the one-hot row x[p, :], recovered as the exact
// f32 weighted sum  Σ_v x[p,v] * v  computed on the WMMA pipe
// (V_WMMA_F32_16X16X4_F32, 16 positions per wave in the M dimension,
//  B = broadcast column of k-indices, C accumulates across K=16384).
//
// Memory-bound on streaming x (1 GiB @ 23.3 TB/s ≈ 46 us floor).
// ---------------------------------------------------------------------------

typedef __attribute__((ext_vector_type(2))) float v2f;
typedef __attribute__((ext_vector_type(4))) float v4f;
typedef __attribute__((ext_vector_type(8))) float v8f;

#define V_DIM   16384
#define E_DIM   512
#define S_DIM   64
#define F_DIM   64
#define BSF     16384      // B*S*F positions
#define ROWS    16         // positions per wave (WMMA M)
#define KTILE   128        // K columns staged per LDS tile
#define LDS_STRIDE 132     // padded row stride (floats): 16B-aligned rows,
                           // bank = (4*m + c) % 64 -> conflict-free A reads

// ---------------------------------------------------------------------------
// Kernel 1: tiled transpose W[E,V] -> Wt[V,E] so gathered columns become
// contiguous 2KB rows (cuts epilogue L2 traffic ~32x vs strided gather).
// ---------------------------------------------------------------------------
__global__ __launch_bounds__(256) void w_transpose_kernel(
    const float* __restrict__ W, float* __restrict__ Wt) {
  __shared__ float tile[32][33];
  const int tx = threadIdx.x;              // 0..31 (lane within row)
  const int ty = threadIdx.y;              // 0..7
  const int v0 = blockIdx.x * 32;
  const int e0 = blockIdx.y * 32;
#pragma unroll
  for (int i = 0; i < 32; i += 8)
    tile[ty + i][tx] = W[(size_t)(e0 + ty + i) * V_DIM + (v0 + tx)];
  __syncthreads();
#pragma unroll
  for (int i = 0; i < 32; i += 8)
    Wt[(size_t)(v0 + ty + i) * E_DIM + (e0 + tx)] = tile[tx][ty + i];
}

// ---------------------------------------------------------------------------
// Kernel 2: one wave (32 threads) per 16 positions.
//   Phase 1: stream x through LDS, WMMA-accumulate idx = sum(x * v).
//   Phase 2: readlane indices, gather W (+pos +fmap), write out.
// ---------------------------------------------------------------------------
__global__ __launch_bounds__(32) void onehot_embed_kernel(
    const float* __restrict__ x,      // [BSF, V] one-hot rows
    const float* __restrict__ Wp,     // Wt[V,E] if use_wt, else W[E,V]
    const float* __restrict__ pos,    // [256, E]
    const float* __restrict__ fmap,   // [256, E]
    float* __restrict__ out,          // [BSF, E]
    int use_wt) {
  __shared__ float lds[ROWS * LDS_STRIDE];

  const int lane    = threadIdx.x;         // 0..31
  const int m16     = lane & 15;           // A-matrix row this lane feeds
  const int halfSel = lane >> 4;           // 0: K0,K1   1: K2,K3
  const size_t rbase = (size_t)blockIdx.x * ROWS;
  const float* xbase = x + rbase * (size_t)V_DIM;

  v8f acc = {};  // f32 C/D accumulator: D[m][n] = sum_k x[m,k] * k (all n equal)

  for (int kbase = 0; kbase < V_DIM; kbase += KTILE) {
    // ---- stage tile: 16 rows x 128 floats; each wave-iteration loads one
    //      row as 32 lanes x b128 = 512 contiguous bytes ----
#pragma unroll
    for (int i = 0; i < ROWS; ++i) {
      v4f d = *(const v4f*)(xbase + (size_t)i * V_DIM + kbase + lane * 4);
      *(v4f*)(&lds[i * LDS_STRIDE + lane * 4]) = d;
    }
    __syncthreads();

    // ---- 32 WMMA K-steps over the tile ----
#pragma unroll
    for (int k4 = 0; k4 < KTILE / 4; ++k4) {
      // A fragment: lanes 0-15 carry {K0,K1} of row m, lanes 16-31 {K2,K3}
      v2f a = *(const v2f*)(&lds[m16 * LDS_STRIDE + k4 * 4 + halfSel * 2]);
      // B fragment: B[k][n] = global k index (same for every column n)
      float kg = (float)(kbase + k4 * 4 + halfSel * 2);
      v2f b;  b.x = kg;  b.y = kg + 1.0f;
      acc = __builtin_amdgcn_wmma_f32_16x16x4_f32(
          /*neg_a=*/false, a, /*neg_b=*/false, b,
          /*c_mod=*/(short)0, acc, /*reuse_a=*/false, /*reuse_b=*/false);
    }
    __syncthreads();  // protect LDS tile against next iteration's stores
  }

  // ---- extract the 16 indices: row m lives in C-VGPR (m%8),
  //      lanes 0..15 for m<8, lanes 16..31 for m>=8 ----
  int idx[ROWS];
#pragma unroll
  for (int m = 0; m < 8; ++m) {
    int lo = __builtin_amdgcn_readlane(__builtin_bit_cast(int, acc[m]), 0);
    int hi = __builtin_amdgcn_readlane(__builtin_bit_cast(int, acc[m]), 16);
    idx[m]     = (int)(__builtin_bit_cast(float, lo) + 0.5f);
    idx[m + 8] = (int)(__builtin_bit_cast(float, hi) + 0.5f);
  }

  // ---- gather + bias epilogue ----
#pragma unroll 1
  for (int m = 0; m < ROWS; ++m) {
    const size_t p = rbase + m;
    const int s = (int)((p >> 6) & 63);
    const int f = (int)(p & 63);
    const float* posrow  = pos  + (size_t)s * E_DIM;
    const float* fmaprow = fmap + (size_t)f * E_DIM;
    float* outrow = out + p * (size_t)E_DIM;
    const int id = idx[m];

    if (use_wt) {
      // contiguous 2KB row of Wt: fully coalesced b128 loads/stores
      const float* wrow = Wp + (size_t)id * E_DIM;
#pragma unroll
      for (int e4 = lane; e4 < E_DIM / 4; e4 += 32) {
        v4f w = *(const v4f*)(wrow + e4 * 4);
        v4f pr = *(const v4f*)(posrow + e4 * 4);
        v4f fr = *(const v4f*)(fmaprow + e4 * 4);
        v4f o;
        o.x = w.x + pr.x + fr.x;  o.y = w.y + pr.y + fr.y;
        o.z = w.z + pr.z + fr.z;  o.w = w.w + pr.w + fr.w;
        *(v4f*)(outrow + e4 * 4) = o;
      }
    } else {
      // fallback: strided column gather from L2-resident W[E,V]
      for (int e = lane; e < E_DIM; e += 32)
        outrow[e] = Wp[(size_t)e * V_DIM + id] + posrow[e] + fmaprow[e];
    }
  }
}

// ---------------------------------------------------------------------------
extern "C" void kernel_launch(void* const* d_in, const int* in_sizes, int n_in,
                              void* d_out, int out_size, void* d_ws, size_t ws_size,
                              hipStream_t stream) {
  const float* x    = (const float*)d_in[0];  // [4,64,64,16384] f32 one-hot
  const float* W    = (const float*)d_in[1];  // [512,16384] f32
  const float* pos  = (const float*)d_in[2];  // [256,512] f32
  const float* fmap = (const float*)d_in[3];  // [256,512] f32
  float* out = (float*)d_out;                 // [4,64,64,512] f32

  const size_t wt_bytes = (size_t)V_DIM * E_DIM * sizeof(float);  // 32 MB
  const bool use_wt = (ws_size >= wt_bytes);
  float* Wt = (float*)d_ws;

  if (use_wt) {
    dim3 tb(32, 8);
    dim3 tg(V_DIM / 32, E_DIM / 32);  // (512, 16)
    w_transpose_kernel<<<tg, tb, 0, stream>>>(W, Wt);
  }

  const int nblocks = BSF / ROWS;  // 1024 single-wave workgroups
  onehot_embed_kernel<<<nblocks, 32, 0, stream>>>(
      x, use_wt ? Wt : W, pos, fmap, out, use_wt ? 1 : 0);
}